// DetectionLoss_39127152066864
// MI455X (gfx1250) — compile-verified
//
#include <hip/hip_runtime.h>
#include <math.h>

#define BLOCK 256
#define CHUNKS 2                       // float4 chunks per thread per array per stage
#define STAGE_F4 (BLOCK * CHUNKS)      // 512 float4 per array per stage
#define STAGE_ELEMS (STAGE_F4 * 4)     // 2048 floats per array per stage
#define STAGE_BYTES (STAGE_ELEMS * 4)  // 8192 bytes per array per stage

// Async copy of 16 bytes/lane from global (SGPR base + VGPR byte offset, GVS mode)
// into LDS at a per-lane 32-bit LDS byte offset. Tracked by ASYNCcnt.
__device__ __forceinline__ void async_cp16(unsigned ldsOff, unsigned long long gbase,
                                           unsigned gOff) {
  asm volatile("global_load_async_to_lds_b128 %0, %1, %2"
               :: "v"(ldsOff), "v"(gOff), "s"(gbase)
               : "memory");
}

__global__ __launch_bounds__(BLOCK) void detloss_reduce(
    const float* __restrict__ outs, const float* __restrict__ labs, long long N,
    float* __restrict__ wsq, unsigned* __restrict__ wtp, unsigned* __restrict__ wfn) {
  // Double-buffered stage tiles (16 KB + 16 KB = 32 KB of the 320 KB/WGP LDS).
  __shared__ float4 s_out[2 * STAGE_F4];
  __shared__ float4 s_lab[2 * STAGE_F4];
  __shared__ float    red_sq[BLOCK / 32];
  __shared__ unsigned red_tp[BLOCK / 32];
  __shared__ unsigned red_fn[BLOCK / 32];

  const int t = threadIdx.x;
  const unsigned ldsOutBase = (unsigned)(size_t)(&s_out[0]);  // low 32 bits = LDS offset
  const unsigned ldsLabBase = (unsigned)(size_t)(&s_lab[0]);
  const unsigned long long outB = (unsigned long long)outs;
  const unsigned long long labB = (unsigned long long)labs;

  float sq = 0.0f;
  unsigned tp = 0u, fn = 0u;

  const long long nStages = N / STAGE_ELEMS;
  long long s = blockIdx.x;
  int buf = 0;

  if (s < nStages) {
    // Prologue: issue stage s into buffer 0 (4 async b128 ops per wave-lane pass).
    {
      unsigned gb = (unsigned)(s * STAGE_BYTES);
#pragma unroll
      for (int j = 0; j < CHUNKS; ++j) {
        unsigned idx  = (unsigned)(j * BLOCK + t);
        unsigned loff = idx * 16u;
        unsigned goff = gb + idx * 16u;
        async_cp16(ldsOutBase + loff, outB, goff);
        async_cp16(ldsLabBase + loff, labB, goff);
      }
    }
    while (true) {
      long long snext = s + (long long)gridDim.x;
      bool hasNext = snext < nStages;
      if (hasNext) {
        // Issue next stage into the other buffer, then wait for the previous 4
        // async ops (async loads complete in order -> asynccnt<=4 means the
        // current buffer is fully resident in LDS).
        unsigned gb    = (unsigned)(snext * STAGE_BYTES);
        unsigned lbase = (unsigned)((buf ^ 1) * STAGE_F4) * 16u;
#pragma unroll
        for (int j = 0; j < CHUNKS; ++j) {
          unsigned idx  = (unsigned)(j * BLOCK + t);
          unsigned loff = lbase + idx * 16u;
          unsigned goff = gb + idx * 16u;
          async_cp16(ldsOutBase + loff, outB, goff);
          async_cp16(ldsLabBase + loff, labB, goff);
        }
        asm volatile("s_wait_asynccnt 0x4" ::: "memory");
      } else {
        asm volatile("s_wait_asynccnt 0x0" ::: "memory");
      }
      // Consume current buffer (each lane reads back exactly the slots it
      // requested -> no cross-wave sharing, no barrier needed in the hot loop).
#pragma unroll
      for (int j = 0; j < CHUNKS; ++j) {
        int idx = buf * STAGE_F4 + j * BLOCK + t;
        float4 o = s_out[idx];
        float4 l = s_lab[idx];
        { float d = o.x - l.x; sq = fmaf(d, d, sq);
          bool p = l.x > 0.5f, q = o.x > 0.5f;
          tp += (unsigned)(p && q); fn += (unsigned)(p && !q); }
        { float d = o.y - l.y; sq = fmaf(d, d, sq);
          bool p = l.y > 0.5f, q = o.y > 0.5f;
          tp += (unsigned)(p && q); fn += (unsigned)(p && !q); }
        { float d = o.z - l.z; sq = fmaf(d, d, sq);
          bool p = l.z > 0.5f, q = o.z > 0.5f;
          tp += (unsigned)(p && q); fn += (unsigned)(p && !q); }
        { float d = o.w - l.w; sq = fmaf(d, d, sq);
          bool p = l.w > 0.5f, q = o.w > 0.5f;
          tp += (unsigned)(p && q); fn += (unsigned)(p && !q); }
      }
      if (!hasNext) break;
      s = snext;
      buf ^= 1;
    }
  }

  // Tail elements (N not a multiple of STAGE_ELEMS): direct loads.
  for (long long i = nStages * STAGE_ELEMS + (long long)blockIdx.x * BLOCK + t;
       i < N; i += (long long)gridDim.x * BLOCK) {
    float o = outs[i], l = labs[i];
    float d = o - l; sq = fmaf(d, d, sq);
    bool p = l > 0.5f, q = o > 0.5f;
    tp += (unsigned)(p && q); fn += (unsigned)(p && !q);
  }

  // Wave32 reduction (gfx1250 is wave32-only).
#pragma unroll
  for (int off = 16; off > 0; off >>= 1) {
    sq += __shfl_xor(sq, off, 32);
    tp += (unsigned)__shfl_xor((int)tp, off, 32);
    fn += (unsigned)__shfl_xor((int)fn, off, 32);
  }
  const int wid = t >> 5, lane = t & 31;
  if (lane == 0) { red_sq[wid] = sq; red_tp[wid] = tp; red_fn[wid] = fn; }
  __syncthreads();
  if (t == 0) {
    float S = 0.0f; unsigned T = 0u, F = 0u;
#pragma unroll
    for (int i = 0; i < BLOCK / 32; ++i) { S += red_sq[i]; T += red_tp[i]; F += red_fn[i]; }
    wsq[blockIdx.x] = S; wtp[blockIdx.x] = T; wfn[blockIdx.x] = F;
  }
}

__global__ __launch_bounds__(BLOCK) void detloss_finalize(
    const float* __restrict__ wsq, const unsigned* __restrict__ wtp,
    const unsigned* __restrict__ wfn, int nb, float* __restrict__ out, long long N) {
  __shared__ double             rs[BLOCK];
  __shared__ unsigned long long rt[BLOCK];
  __shared__ unsigned long long rf[BLOCK];
  const int t = threadIdx.x;
  double sq = 0.0;
  unsigned long long tp = 0ull, fn = 0ull;
  for (int i = t; i < nb; i += BLOCK) {
    sq += (double)wsq[i];
    tp += (unsigned long long)wtp[i];
    fn += (unsigned long long)wfn[i];
  }
  rs[t] = sq; rt[t] = tp; rf[t] = fn;
  __syncthreads();
  for (int off = BLOCK / 2; off > 0; off >>= 1) {
    if (t < off) { rs[t] += rs[t + off]; rt[t] += rt[t + off]; rf[t] += rf[t + off]; }
    __syncthreads();
  }
  if (t == 0) {
    float mse = (float)(rs[0] / (double)N);
    unsigned long long TP = rt[0], FN = rf[0];
    float coeff;
    if (TP == 0ull && FN == 0ull)      coeff = 1.0f;
    else if (TP == 0ull)               coeff = 0.0f;
    else                               coeff = (float)((double)TP / (double)(TP + FN));
    float cost = -logf(coeff + 1e-10f);
    out[0] = mse + 0.5f * cost;
  }
}

extern "C" void kernel_launch(void* const* d_in, const int* in_sizes, int n_in,
                              void* d_out, int out_size, void* d_ws, size_t ws_size,
                              hipStream_t stream) {
  const float* outputs = (const float*)d_in[0];
  const float* labels  = (const float*)d_in[1];
  long long N = (long long)in_sizes[0];

  int nb = 2048;
  long long stages = (N + STAGE_ELEMS - 1) / STAGE_ELEMS;
  if (stages > 0 && stages < (long long)nb) nb = (int)stages;
  size_t maxnb = ws_size / 12;  // float + uint + uint per block
  if ((size_t)nb > maxnb) nb = (int)maxnb;
  if (nb < 1) nb = 1;

  float*    wsq = (float*)d_ws;
  unsigned* wtp = (unsigned*)(wsq + nb);
  unsigned* wfn = wtp + nb;

  detloss_reduce<<<nb, BLOCK, 0, stream>>>(outputs, labels, N, wsq, wtp, wfn);
  detloss_finalize<<<1, BLOCK, 0, stream>>>(wsq, wtp, wfn, nb, (float*)d_out, N);
}